// Deform_conv3d_90125593739962
// MI455X (gfx1250) — compile-verified
//
#include <hip/hip_runtime.h>
#include <math.h>

#define DHW (64 * 64 * 64)

typedef __attribute__((ext_vector_type(2))) float v2f;
typedef __attribute__((ext_vector_type(8))) float v8f;
typedef __attribute__((ext_vector_type(4))) unsigned int v4u;
typedef __attribute__((ext_vector_type(4))) int v4i;
typedef __attribute__((ext_vector_type(8))) int v8i;

// ---------------------------------------------------------------------------
// Kernel A: 3-channel 5x5x5 offset conv + tanh, fused with grid construction.
// Emits the final trilinear sample coordinates (ix, iy, iz) per voxel.
// ---------------------------------------------------------------------------
__global__ __launch_bounds__(256) void kern_offsets(
    const float* __restrict__ x,       // [N,32,64,64,64]
    const float* __restrict__ w_def,   // [3,32,5,5,5]
    const float* __restrict__ b_def,   // [3]
    float* __restrict__ coords)        // [N*DHW, 3] -> (ix, iy, iz)
{
    __shared__ float wl[3 * 32 * 125];  // 48 KB
    for (int i = threadIdx.x; i < 3 * 32 * 125; i += 256) wl[i] = w_def[i];
    __syncthreads();

    int gid = blockIdx.x * 256 + threadIdx.x;   // [0, N*DHW)
    int n = gid / DHW;
    int v = gid % DHW;
    int d = v >> 12;
    int h = (v >> 6) & 63;
    int w = v & 63;

    const float* xb = x + (size_t)n * 32 * DHW;
    float a0 = b_def[0], a1 = b_def[1], a2 = b_def[2];

    for (int c = 0; c < 32; ++c) {
        const float* xc = xb + (size_t)c * DHW;
        const float* wc = wl + c * 125;
        for (int kd = 0; kd < 5; ++kd) {
            int zz = d + kd - 2;
            if (zz < 0 || zz > 63) continue;
            for (int kh = 0; kh < 5; ++kh) {
                int yy = h + kh - 2;
                if (yy < 0 || yy > 63) continue;
                int rowbase = (zz << 12) + (yy << 6);
                int wbase = kd * 25 + kh * 5;
#pragma unroll
                for (int kw = 0; kw < 5; ++kw) {
                    int xx = w + kw - 2;
                    if (xx < 0 || xx > 63) continue;
                    float xv = xc[rowbase + xx];
                    a0 = fmaf(wc[wbase + kw], xv, a0);
                    a1 = fmaf(wc[4000 + wbase + kw], xv, a1);
                    a2 = fmaf(wc[8000 + wbase + kw], xv, a2);
                }
            }
        }
    }

    // grid = regular_grid + tanh(off)/[D,H,W]; then torch-style unnormalize
    const float step = 2.0f / 63.0f;
    float g0 = -1.0f + d * step + tanhf(a0) * (1.0f / 64.0f);  // -> ix (W axis)
    float g1 = -1.0f + h * step + tanhf(a1) * (1.0f / 64.0f);  // -> iy (H axis)
    float g2 = -1.0f + w * step + tanhf(a2) * (1.0f / 64.0f);  // -> iz (D axis)
    float ix = (g0 + 1.0f) * 32.0f - 0.5f;
    float iy = (g1 + 1.0f) * 32.0f - 0.5f;
    float iz = (g2 + 1.0f) * 32.0f - 0.5f;

    float* cp = coords + (size_t)gid * 3;
    cp[0] = ix;
    cp[1] = iy;
    cp[2] = iz;
}

// ---------------------------------------------------------------------------
// Kernel B: trilinear grid-sample (zeros padding). One thread per voxel,
// corner weights/indices computed once, 32-channel loop. Output is written
// CHANNEL-LAST ([n][voxel][32]) so the WMMA consumer can fetch its B-operand
// K-slices with wide contiguous loads (one 128-B record per voxel).
// ---------------------------------------------------------------------------
__global__ __launch_bounds__(256) void kern_sample(
    const float* __restrict__ x,       // [N,32,64,64,64]
    const float* __restrict__ coords,  // [N*DHW, 3]
    float* __restrict__ xot)           // [N, DHW, 32] channel-last
{
    int gid = blockIdx.x * 256 + threadIdx.x;
    int n = gid / DHW;
    int v = gid % DHW;

    const float* cp = coords + (size_t)gid * 3;
    float ix = cp[0], iy = cp[1], iz = cp[2];
    float fx0 = floorf(ix), fy0 = floorf(iy), fz0 = floorf(iz);
    float fx = ix - fx0, fy = iy - fy0, fz = iz - fz0;
    int x0 = (int)fx0, y0 = (int)fy0, z0 = (int)fz0;

    int lin[8];
    float wgt[8];
#pragma unroll
    for (int k = 0; k < 8; ++k) {
        int dz = k >> 2, dy = (k >> 1) & 1, dx = k & 1;
        int zc = z0 + dz, yc = y0 + dy, xc = x0 + dx;
        bool valid = (zc >= 0) && (zc < 64) && (yc >= 0) && (yc < 64) &&
                     (xc >= 0) && (xc < 64);
        float wv = (dz ? fz : 1.0f - fz) * (dy ? fy : 1.0f - fy) *
                   (dx ? fx : 1.0f - fx);
        wgt[k] = valid ? wv : 0.0f;
        int zcc = min(max(zc, 0), 63);
        int ycc = min(max(yc, 0), 63);
        int xcc = min(max(xc, 0), 63);
        lin[k] = (zcc << 12) + (ycc << 6) + xcc;
    }

    const float* xb = x + (size_t)n * 32 * DHW;
    float acc[32];
    for (int c = 0; c < 32; ++c) {
        const float* xc = xb + (size_t)c * DHW;
        float a = 0.0f;
#pragma unroll
        for (int k = 0; k < 8; ++k) a = fmaf(wgt[k], xc[lin[k]], a);
        acc[c] = a;
    }

    float4* op = (float4*)(xot + (size_t)gid * 32);
#pragma unroll
    for (int j = 0; j < 8; ++j)
        op[j] = make_float4(acc[4 * j], acc[4 * j + 1], acc[4 * j + 2],
                            acc[4 * j + 3]);
}

// ---------------------------------------------------------------------------
// Kernel C: 32->32 3x3x3 conv as implicit GEMM on V_WMMA_F32_16X16X4_F32.
// M = 32 output channels (two 16x16 accumulators), N = 16 voxels per wave,
// K = 864 = 27 taps x 32 channels, stepped 4 channels per WMMA issue.
//
// f32 WMMA lane layout (wave32, hi = lane>>4, l16 = lane&15):
//   A 16x4 : lane holds A[M=l16][K = 2*hi + {0,1}]
//   B  4x16: lane holds B[K = 2*hi + {0,1}][N = l16]
//   C/D    : VGPR v holds row M = v + 8*hi, col N = l16
//
// Weights (110.6 KB) are staged into LDS by the Tensor Data Mover:
// one tensor_load_to_lds per block (wave 0), s_wait_tensorcnt 0, barrier.
// ---------------------------------------------------------------------------
__global__ __launch_bounds__(256) void kern_conv2_wmma(
    const float* __restrict__ xot,     // [N, DHW, 32] channel-last
    const float* __restrict__ w_conv,  // [32,32,3,3,3] = [oc][c][tap]
    const float* __restrict__ b_conv,  // [32]
    float* __restrict__ out)           // [N,32,64,64,64]
{
    __shared__ float wl[32 * 864];  // 110.6 KB of the 320 KB WGP LDS

    if (threadIdx.x == 0) {
        // Tensor DMA descriptor (CDNA5 ISA):
        // 1-D tile: 27648 x f32 from w_conv -> LDS offset 0.
        unsigned long long ga = (unsigned long long)(uintptr_t)w_conv;
        v4u g0;
        g0.x = 1u;                                   // count=1 (valid D#)
        g0.y = 0u;                                   // lds_addr = 0
        g0.z = (unsigned int)(ga & 0xFFFFFFFFu);     // global_addr[31:0]
        g0.w = (unsigned int)((ga >> 32) & 0x01FFFFFFu) | (2u << 30);  // type=2
        v8i g1;
        g1[0] = 0x00020000;   // workgroup_mask=0, data_size=2 (4 bytes)
        g1[1] = 0x6C000000;   // tensor_dim0[15:0]=27648 in bits[31:16]
        g1[2] = 0x00010000;   // tensor_dim0[31:16]=0, tensor_dim1=1
        g1[3] = 0x6C000000;   // tile_dim0 = 27648 in bits[31:16]
        g1[4] = 1;            // tile_dim1 = 1, tile_dim2 = 0
        g1[5] = 27648;        // tensor_dim0_stride[31:0]
        g1[6] = 0;            // stride high bits / dim1_stride low
        g1[7] = 0;
        v4i g2 = {0, 0, 0, 0};
        v4i g3 = {0, 0, 0, 0};
        v8i g4 = {0, 0, 0, 0, 0, 0, 0, 0};  // extra group (clang-23 6-arg form)
        __builtin_amdgcn_tensor_load_to_lds(g0, g1, g2, g3, g4, 0);
        __builtin_amdgcn_s_wait_tensorcnt(0);
    }
    // The TDM engine wrote wl behind the compiler's back: escape wl's address
    // with a memory clobber so loads from it are not folded to undef.
    asm volatile("" : : "r"(&wl[0]) : "memory");
    __syncthreads();

    int bid = blockIdx.x;            // n*2048 + d*32 + hp
    int n = bid >> 11;
    int d = (bid >> 5) & 63;
    int hp = bid & 31;
    int wave = threadIdx.x >> 5;     // 8 waves per block
    int lane = threadIdx.x & 31;
    int l16 = lane & 15;
    int hi = lane >> 4;
    int h = hp * 2 + (wave >> 2);    // 2 rows per block
    int wx0 = (wave & 3) * 16;       // 4 x-strips per row
    int xpos = wx0 + l16;

    // Bias in the C-matrix layout
    v8f acc0, acc1;
#pragma unroll
    for (int v = 0; v < 8; ++v) {
        acc0[v] = b_conv[v + hi * 8];
        acc1[v] = b_conv[16 + v + hi * 8];
    }

    const v2f* bbase = (const v2f*)(xot + (size_t)n * DHW * 32);

    for (int t = 0; t < 27; ++t) {
        int kd = t / 9, kh = (t / 3) % 3, kw = t % 3;
        int zz = d + kd - 1;
        int yy = h + kh - 1;
        int xx = xpos + kw - 1;
        bool ok = (zz >= 0) && (zz < 64) && (yy >= 0) && (yy < 64) &&
                  (xx >= 0) && (xx < 64);
        float m = ok ? 1.0f : 0.0f;           // arithmetic mask: EXEC stays all-1s
        int zzc = min(max(zz, 0), 63);
        int yyc = min(max(yy, 0), 63);
        int xxc = min(max(xx, 0), 63);
        int spat = (zzc << 12) + (yyc << 6) + xxc;
        size_t sb = (size_t)spat * 16 + hi;   // v2f index of this lane's K pair

#pragma unroll
        for (int cb = 0; cb < 32; cb += 4) {
            v2f b = bbase[sb + (cb >> 1)];    // channels cb+2*hi, cb+2*hi+1
            b.x *= m;
            b.y *= m;

            int c0 = cb + hi * 2;
            int wi = c0 * 27 + t;
            v2f a0, a1;
            a0.x = wl[l16 * 864 + wi];
            a0.y = wl[l16 * 864 + wi + 27];
            a1.x = wl[(l16 + 16) * 864 + wi];
            a1.y = wl[(l16 + 16) * 864 + wi + 27];

            acc0 = __builtin_amdgcn_wmma_f32_16x16x4_f32(
                false, a0, false, b, (short)0, acc0, false, false);
            acc1 = __builtin_amdgcn_wmma_f32_16x16x4_f32(
                false, a1, false, b, (short)0, acc1, false, false);
        }
    }

    int vox = (d << 12) + (h << 6) + xpos;
#pragma unroll
    for (int v = 0; v < 8; ++v) {
        int oc0 = v + hi * 8;
        int oc1 = 16 + v + hi * 8;
        out[(size_t)(n * 32 + oc0) * DHW + vox] = acc0[v];
        out[(size_t)(n * 32 + oc1) * DHW + vox] = acc1[v];
    }
}

// ---------------------------------------------------------------------------
extern "C" void kernel_launch(void* const* d_in, const int* in_sizes, int n_in,
                              void* d_out, int out_size, void* d_ws,
                              size_t ws_size, hipStream_t stream) {
    const float* x      = (const float*)d_in[0];   // [N,32,64,64,64]
    const float* w_def  = (const float*)d_in[1];   // [3,32,5,5,5]
    const float* b_def  = (const float*)d_in[2];   // [3]
    const float* w_conv = (const float*)d_in[3];   // [32,32,3,3,3]
    const float* b_conv = (const float*)d_in[4];   // [32]
    float* out = (float*)d_out;

    int N = in_sizes[0] / (32 * DHW);              // = 2

    // workspace: channel-last x_offset (N*DHW*32 floats) then coords
    float* xot = (float*)d_ws;
    float* coords = xot + (size_t)N * DHW * 32;

    int total = N * DHW;
    kern_offsets<<<total / 256, 256, 0, stream>>>(x, w_def, b_def, coords);
    kern_sample<<<total / 256, 256, 0, stream>>>(x, coords, xot);
    kern_conv2_wmma<<<N * 64 * 32, 256, 0, stream>>>(xot, w_conv, b_conv, out);
}